// PhraseClassifier_43997644981009
// MI455X (gfx1250) — compile-verified
//
#include <hip/hip_runtime.h>
#include <hip/hip_bf16.h>
#include <cmath>

typedef __attribute__((ext_vector_type(2))) float v2f;
typedef __attribute__((ext_vector_type(4))) float v4f;
typedef __attribute__((ext_vector_type(8))) float v8f;

#define HDIM        512
#define TWO_H       1024
#define BDIM        32
#define SPB         32            // spans per block (2 row tiles of WMMA M=16)
#define FEAT_STRIDE 1028          // 1024 + 4 floats pad -> bank stride 4 mod 64
#define NWAVES      16
#define THREADS     512

// -------------------- main fused kernel --------------------
__global__ __launch_bounds__(THREADS)
void span_bce_kernel(const float* __restrict__ hidden,
                     const int*   __restrict__ bids,
                     const int*   __restrict__ begins,
                     const int*   __restrict__ ends,
                     const int*   __restrict__ flags,
                     const float* __restrict__ weights,
                     const float* __restrict__ W1,
                     const float* __restrict__ b1,
                     const float* __restrict__ W2,
                     const float* __restrict__ b2,
                     float* __restrict__ part_pos,
                     float* __restrict__ part_neg,
                     float* __restrict__ part_npos,
                     int N)
{
    extern __shared__ float smem[];
    float* feats = smem;                       // [SPB][FEAT_STRIDE]
    float* dotl  = smem + SPB * FEAT_STRIDE;   // [NWAVES][SPB]

    __shared__ int   meta_b[SPB], meta_bg[SPB], meta_e[SPB], meta_f[SPB];
    __shared__ float meta_w[SPB];
    __shared__ float red[3 * SPB];

    const int tid  = threadIdx.x;
    const int blk  = blockIdx.x;
    const int base = blk * SPB;

    if (tid < SPB) {
        int i     = base + tid;
        int valid = (i < N);
        int idx   = valid ? i : (N - 1);
        meta_b[tid]  = bids[idx];
        meta_bg[tid] = begins[idx];
        meta_e[tid]  = ends[idx];
        meta_f[tid]  = valid ? flags[idx] : 0;
        meta_w[tid]  = valid ? weights[idx] : 0.0f;
    }
    __syncthreads();

    // ---- gather + diff: build [32 x 1024] feature tile in LDS ----
    // 32 spans x 256 float4 = 8192 float4 over 512 threads -> 16 iterations
    {
        for (int i = 0; i < 16; ++i) {
            const int L = i * THREADS + tid;
            const int m = L >> 8;         // span 0..31
            const int q = L & 255;        // float4 index
            const int k = q * 4;          // 0..1020
            const int b  = meta_b[m];
            const int bg = meta_bg[m];
            const int e  = meta_e[m];
            v4f val;
            if (k < HDIM) {
                // fwd half: hidden[e-1,b,k] - hidden[bg-1,b,k]
                const float* pa = hidden + ((size_t)(e  - 1) * BDIM + b) * TWO_H + k;
                const float* pb = hidden + ((size_t)(bg - 1) * BDIM + b) * TWO_H + k;
                val = *(const v4f*)pa - *(const v4f*)pb;
            } else {
                // bwd half: hidden[bg,b,H+k'] - hidden[e,b,H+k']
                const float* pa = hidden + ((size_t)bg * BDIM + b) * TWO_H + k;
                const float* pb = hidden + ((size_t)e  * BDIM + b) * TWO_H + k;
                val = *(const v4f*)pa - *(const v4f*)pb;
            }
            *(v4f*)&feats[m * FEAT_STRIDE + k] = val;
        }
    }
    __syncthreads();

    const int wave = tid >> 5;     // wave32: 0..15
    const int lane = tid & 31;
    const int hi   = lane >> 4;    // 0: K pair {0,1}; 1: K pair {2,3}
    const int lo   = lane & 15;    // row (A) / col (B,C)

    // each wave owns column tiles {2*wave, 2*wave+1} x row tiles {0,1}
    const int c0 = wave * 32 + lo;       // col of tile ct0
    const int c1 = c0 + 16;              // col of tile ct1
    const float* w1c0 = W1 + c0;         // row-major [1024][512]
    const float* w1c1 = W1 + c1;

    v8f acc00 = {}, acc01 = {}, acc10 = {}, acc11 = {};

    #pragma unroll 4
    for (int k0 = 0; k0 < TWO_H; k0 += 4) {
        const int kk = k0 + 2 * hi;
        // A fragments (16x4 f32): lane lo = row, VGPR0/1 = K kk,kk+1
        v2f a0 = *(const v2f*)&feats[(lo)      * FEAT_STRIDE + kk];  // rows 0..15
        v2f a1 = *(const v2f*)&feats[(16 + lo) * FEAT_STRIDE + kk];  // rows 16..31
        // B fragments (4x16 f32): lane lo = col, VGPR0/1 = K kk,kk+1
        v2f b0, b1v_;
        b0.x   = w1c0[(size_t)kk       * HDIM];
        b0.y   = w1c0[(size_t)(kk + 1) * HDIM];
        b1v_.x = w1c1[(size_t)kk       * HDIM];
        b1v_.y = w1c1[(size_t)(kk + 1) * HDIM];
        acc00 = __builtin_amdgcn_wmma_f32_16x16x4_f32(false, a0, false, b0,
                                                      (short)0, acc00, false, false);
        acc01 = __builtin_amdgcn_wmma_f32_16x16x4_f32(false, a0, false, b1v_,
                                                      (short)0, acc01, false, false);
        acc10 = __builtin_amdgcn_wmma_f32_16x16x4_f32(false, a1, false, b0,
                                                      (short)0, acc10, false, false);
        acc11 = __builtin_amdgcn_wmma_f32_16x16x4_f32(false, a1, false, b1v_,
                                                      (short)0, acc11, false, false);
    }

    // fused epilogue: + b1, ReLU, * W2[col], per-row dot partials
    const float w2v0 = W2[c0], w2v1 = W2[c1];
    const float b1v0 = b1[c0], b1v1 = b1[c1];

    float pd0[8], pd1[8];        // row tiles 0 and 1
    #pragma unroll
    for (int r = 0; r < 8; ++r) {
        float z;
        z = acc00[r] + b1v0; z = z > 0.0f ? z : 0.0f; pd0[r]  = z * w2v0;
        z = acc01[r] + b1v1; z = z > 0.0f ? z : 0.0f; pd0[r] += z * w2v1;
        z = acc10[r] + b1v0; z = z > 0.0f ? z : 0.0f; pd1[r]  = z * w2v0;
        z = acc11[r] + b1v1; z = z > 0.0f ? z : 0.0f; pd1[r] += z * w2v1;
    }

    // reduce over the 16 column-lanes of each half-wave
    #pragma unroll
    for (int r = 0; r < 8; ++r) {
        float v0 = pd0[r], v1 = pd1[r];
        for (int off = 1; off < 16; off <<= 1) {
            v0 += __shfl_xor(v0, off, 32);
            v1 += __shfl_xor(v1, off, 32);
        }
        pd0[r] = v0; pd1[r] = v1;
    }
    if (lo == 0) {
        #pragma unroll
        for (int r = 0; r < 8; ++r) {
            const int m = r + 8 * hi;                 // C row within tile
            dotl[wave * SPB + m]      = pd0[r];       // spans 0..15
            dotl[wave * SPB + 16 + m] = pd1[r];       // spans 16..31
        }
    }
    __syncthreads();

    // ---- per-span sigmoid + BCE (threads 0..31), deterministic block sums ----
    if (tid < SPB) {
        float dot = b2[0];
        for (int w = 0; w < NWAVES; ++w) dot += dotl[w * SPB + tid];
        float p  = 1.0f / (1.0f + expf(-dot));
        float pc = fminf(fmaxf(p, 1e-7f), 1.0f - 1e-7f);
        const int   fl = meta_f[tid];
        const float w  = meta_w[tid];
        const float bce = (fl == 1) ? -logf(pc) : -logf(1.0f - pc);
        red[tid]           = (fl == 1) ? w * bce : 0.0f;
        red[SPB + tid]     = (fl == 1) ? 0.0f : w * bce;
        red[2 * SPB + tid] = (fl == 1) ? 1.0f : 0.0f;
    }
    __syncthreads();
    if (tid == 0) {
        float sp = 0.0f, sn = 0.0f, np = 0.0f;
        for (int i = 0; i < SPB; ++i) {
            sp += red[i];
            sn += red[SPB + i];
            np += red[2 * SPB + i];
        }
        part_pos[blk]  = sp;
        part_neg[blk]  = sn;
        part_npos[blk] = np;
    }
}

// -------------------- finalize: deterministic reduction --------------------
__global__ __launch_bounds__(256)
void finalize_kernel(const float* __restrict__ part_pos,
                     const float* __restrict__ part_neg,
                     const float* __restrict__ part_npos,
                     int nblk, int N, float* __restrict__ out)
{
    __shared__ float s0[256], s1[256], s2[256];
    const int tid = threadIdx.x;
    float sp = 0.0f, sn = 0.0f, np = 0.0f;
    for (int i = tid; i < nblk; i += 256) {
        sp += part_pos[i];
        sn += part_neg[i];
        np += part_npos[i];
    }
    s0[tid] = sp; s1[tid] = sn; s2[tid] = np;
    __syncthreads();
    for (int off = 128; off > 0; off >>= 1) {
        if (tid < off) {
            s0[tid] += s0[tid + off];
            s1[tid] += s1[tid + off];
            s2[tid] += s2[tid + off];
        }
        __syncthreads();
    }
    if (tid == 0) {
        const float scale = 2.0f * s2[0] / (float)N;
        out[0] = (s0[0] + scale * s1[0]) / (float)N;
    }
}

// -------------------- launch --------------------
extern "C" void kernel_launch(void* const* d_in, const int* in_sizes, int n_in,
                              void* d_out, int out_size, void* d_ws, size_t ws_size,
                              hipStream_t stream) {
    const float* hidden  = (const float*)d_in[0];
    const int*   bids    = (const int*)  d_in[1];
    const int*   begins  = (const int*)  d_in[2];
    const int*   ends    = (const int*)  d_in[3];
    const int*   flags   = (const int*)  d_in[4];
    const float* weights = (const float*)d_in[5];
    const float* W1      = (const float*)d_in[6];
    const float* b1      = (const float*)d_in[7];
    const float* W2      = (const float*)d_in[8];
    const float* b2      = (const float*)d_in[9];

    const int N    = in_sizes[1];
    const int nblk = (N + SPB - 1) / SPB;

    float* part_pos  = (float*)d_ws;
    float* part_neg  = part_pos + nblk;
    float* part_npos = part_neg + nblk;

    const size_t lds_bytes = (size_t)(SPB * FEAT_STRIDE + NWAVES * SPB) * sizeof(float);

    span_bce_kernel<<<nblk, THREADS, lds_bytes, stream>>>(
        hidden, bids, begins, ends, flags, weights, W1, b1, W2, b2,
        part_pos, part_neg, part_npos, N);

    finalize_kernel<<<1, 256, 0, stream>>>(part_pos, part_neg, part_npos,
                                           nblk, N, (float*)d_out);
}